// UAFlow_68367289418164
// MI455X (gfx1250) — compile-verified
//
#include <hip/hip_runtime.h>

typedef __attribute__((ext_vector_type(2))) float v2f;
typedef __attribute__((ext_vector_type(8))) float v8f;

#define BATCH   64
#define KDIM    196608            // 3*256*256
#define TSCALE  0.99f             // 1 - 1/MAX_TIMESTEPS
#define NBLK1   48
#define KCHUNK1 4096              // KDIM / NBLK1
#define NBLK3   768
#define NTILES_K 12288            // KDIM / 16

// workspace layout (float offsets)
#define CROSS_OFF 0               // 48 * 4096
#define N2_OFF    196608          // 48 * 64
#define D2_OFF    199680          // 48 * 64
#define W_OFF     202752          // 64 * 64
#define LOSS_OFF  206848          // NBLK3
// total ~830 KB

// ---------------- Pass 1: cross = noised @ data^T, plus n2/d2 partials ----
__global__ __launch_bounds__(512) void k1_cross(const float* __restrict__ data,
                                                const float* __restrict__ noise,
                                                const float* __restrict__ times,
                                                float* __restrict__ ws)
{
    const int lane = threadIdx.x & 31;
    const int wave = threadIdx.x >> 5;
    const int ti = wave >> 2;          // i-tile 0..3
    const int tj = wave & 3;           // j-tile 0..3
    const int half = lane >> 4;        // 0: K pair {0,1}, 1: K pair {2,3}
    const int lr   = lane & 15;

    const int ia = ti * 16 + lr;       // noised row for A fragment
    const int jb = tj * 16 + lr;       // data row for B fragment
    const float t_a = times[ia] * TSCALE;
    const long long aBase = (long long)ia * KDIM;
    const long long bBase = (long long)jb * KDIM;

    const int kbeg = blockIdx.x * KCHUNK1;
    v8f c = {};
    float accA = 0.f, accB = 0.f;

    for (int k0 = kbeg; k0 < kbeg + KCHUNK1; k0 += 4) {
        const int kk = k0 + 2 * half;
        // A fragment: noised = noise + t*(data - noise)
        v2f dA = *(const v2f*)(data  + aBase + kk);
        v2f nA = *(const v2f*)(noise + aBase + kk);
        v2f a;
        a.x = nA.x + t_a * (dA.x - nA.x);
        a.y = nA.y + t_a * (dA.y - nA.y);
        // B fragment: B[k, n] = data[j0+n, k]
        v2f b = *(const v2f*)(data + bBase + kk);
        c = __builtin_amdgcn_wmma_f32_16x16x4_f32(false, a, false, b,
                                                  (short)0, c, false, false);
        if (tj == 0) accA += a.x * a.x + a.y * a.y;   // ||noised||^2 partial
        if (ti == 0) accB += b.x * b.x + b.y * b.y;   // ||data||^2 partial
    }

    // store 16x16 cross tile partial (C layout: vgpr r -> row r + 8*half, lane -> col)
    float* cross_part = ws + CROSS_OFF + (long long)blockIdx.x * 4096;
#pragma unroll
    for (int r = 0; r < 8; ++r) {
        int m = ti * 16 + r + 8 * half;
        int n = tj * 16 + lr;
        cross_part[m * 64 + n] = c[r];
    }
    if (tj == 0) {
        float s = accA + __shfl_down(accA, 16, 32);
        if (lane < 16) ws[N2_OFF + blockIdx.x * 64 + ti * 16 + lane] = s;
    }
    if (ti == 0) {
        float s = accB + __shfl_down(accB, 16, 32);
        if (lane < 16) ws[D2_OFF + blockIdx.x * 64 + tj * 16 + lane] = s;
    }
}

// ---------------- Pass 2: reduce partials, dist + softmax -> weights ------
__global__ __launch_bounds__(256) void k2_softmax(const float* __restrict__ times,
                                                  float* __restrict__ ws)
{
    __shared__ float sc[4096];
    __shared__ float sn2[64], sd2[64];

    for (int idx = threadIdx.x; idx < 4096; idx += 256) {
        float s = 0.f;
        for (int p = 0; p < NBLK1; ++p) s += ws[CROSS_OFF + p * 4096 + idx];
        sc[idx] = s;
    }
    if (threadIdx.x < 64) {
        float s = 0.f, s2 = 0.f;
        for (int p = 0; p < NBLK1; ++p) {
            s  += ws[N2_OFF + p * 64 + threadIdx.x];
            s2 += ws[D2_OFF + p * 64 + threadIdx.x];
        }
        sn2[threadIdx.x] = s;
        sd2[threadIdx.x] = s2;
    }
    __syncthreads();

    if (threadIdx.x < 64) {
        const int i = threadIdx.x;
        const float t = times[i] * TSCALE;
        const float var = (1.f - t) * (1.f - t) + 1e-8f;
        const float inv2v = 1.f / (2.f * var);
        float dist[64];
        float dmin = 3.4e38f;
        for (int j = 0; j < 64; ++j) {
            float d = (sn2[i] - 2.f * t * sc[i * 64 + j] + t * t * sd2[j]) * inv2v;
            dist[j] = d;
            dmin = fminf(dmin, d);
        }
        float sum = 0.f;
        for (int j = 0; j < 64; ++j) {
            float e = __expf(-(dist[j] - dmin));   // softmax(-dist)
            dist[j] = e;
            sum += e;
        }
        const float inv = 1.f / sum;
        for (int j = 0; j < 64; ++j) ws[W_OFF + i * 64 + j] = dist[j] * inv;
    }
}

// ---------------- Pass 3: mix = W @ data (WMMA) + fused loss --------------
__global__ __launch_bounds__(512) void k3_loss(const float* __restrict__ data,
                                               const float* __restrict__ noise,
                                               const float* __restrict__ pm,
                                               const float* __restrict__ plv,
                                               const float* __restrict__ times,
                                               float* __restrict__ ws)
{
    const int lane = threadIdx.x & 31;
    const int wave = threadIdx.x >> 5;
    const int ti   = wave & 3;         // i-tile 0..3
    const int kgrp = wave >> 2;        // k-tile group within block
    const int half = lane >> 4;
    const int lr   = lane & 15;
    const int i0   = ti * 16;

    const float* W = ws + W_OFF;

    // Preload the 16 A fragments (weights rows i0..i0+15, reduction dim j=0..63)
    v2f aW[16];
#pragma unroll
    for (int s = 0; s < 16; ++s) {
        int js = 4 * s + 2 * half;
        aW[s].x = W[(i0 + lr) * 64 + js];
        aW[s].y = W[(i0 + lr) * 64 + js + 1];
    }

    // Per-r constants (row i = i0 + r + 8*half)
    float tArr[8], invm[8];
#pragma unroll
    for (int r = 0; r < 8; ++r) {
        float t = times[i0 + r + 8 * half] * TSCALE;
        tArr[r] = t;
        invm[r] = 1.f / (1.f - t + 1e-8f);
    }

    float acc = 0.f;
    const int kstride = gridDim.x * 4;
    for (int kt = blockIdx.x * 4 + kgrp; kt < NTILES_K; kt += kstride) {
        const int k0 = kt * 16;
        v8f c = {};
#pragma unroll
        for (int s = 0; s < 16; ++s) {
            int j0 = 4 * s + 2 * half;
            v2f b;   // B[j, n] = data[j, k0+n]
            b.x = data[(long long)(j0    ) * KDIM + k0 + lr];
            b.y = data[(long long)(j0 + 1) * KDIM + k0 + lr];
            c = __builtin_amdgcn_wmma_f32_16x16x4_f32(false, aW[s], false, b,
                                                      (short)0, c, false, false);
        }
        // c now holds mix[i0 + r + 8*half, k0 + lr] in vgpr r
#pragma unroll
        for (int r = 0; r < 8; ++r) {
            const int i = i0 + r + 8 * half;
            const long long idx = (long long)i * KDIM + k0 + lr;
            float d  = data[idx];
            float n  = noise[idx];
            float m  = pm[idx];
            float lv = plv[idx];
            float fc = d - n;                        // flow_cond
            float ns = n + tArr[r] * fc;             // noised
            float u  = (c[r] - ns) * invm[r];        // u_hat
            float pv = __expf(lv);
            float e  = m - fc;
            // pred_var * [ (pm-fc)^2/(2pv) + plv/2 + (u^2-fc^2)/(2pv) ]
            acc += 0.5f * (e * e + pv * lv + u * u - fc * fc);
        }
    }

    // wave reduce (wave32), then block reduce
    for (int off = 16; off > 0; off >>= 1) acc += __shfl_down(acc, off, 32);
    __shared__ float sred[16];
    if (lane == 0) sred[wave] = acc;
    __syncthreads();
    if (threadIdx.x == 0) {
        float s = 0.f;
        for (int wv = 0; wv < 16; ++wv) s += sred[wv];
        ws[LOSS_OFF + blockIdx.x] = s;
    }
}

// ---------------- Pass 4: final scalar --------------------------------------
__global__ __launch_bounds__(256) void k4_final(const float* __restrict__ ws,
                                                float* __restrict__ out)
{
    __shared__ float s[256];
    float acc = 0.f;
    for (int i = threadIdx.x; i < NBLK3; i += 256) acc += ws[LOSS_OFF + i];
    s[threadIdx.x] = acc;
    __syncthreads();
    for (int off = 128; off > 0; off >>= 1) {
        if (threadIdx.x < off) s[threadIdx.x] += s[threadIdx.x + off];
        __syncthreads();
    }
    if (threadIdx.x == 0)
        out[0] = s[0] / 12582912.0f;   // mean over 64*196608 elements
}

extern "C" void kernel_launch(void* const* d_in, const int* in_sizes, int n_in,
                              void* d_out, int out_size, void* d_ws, size_t ws_size,
                              hipStream_t stream)
{
    const float* data  = (const float*)d_in[0];
    const float* noise = (const float*)d_in[1];
    const float* times = (const float*)d_in[2];
    const float* pmean = (const float*)d_in[3];
    const float* plvar = (const float*)d_in[4];
    float* out = (float*)d_out;
    float* ws  = (float*)d_ws;

    k1_cross  <<<NBLK1, 512, 0, stream>>>(data, noise, times, ws);
    k2_softmax<<<1,     256, 0, stream>>>(times, ws);
    k3_loss   <<<NBLK3, 512, 0, stream>>>(data, noise, pmean, plvar, times, ws);
    k4_final  <<<1,     256, 0, stream>>>(ws, out);
}